// ComplexSVDTransform_6932077216155
// MI455X (gfx1250) — compile-verified
//
#include <hip/hip_runtime.h>
#include <hip/hip_bf16.h>
#include <math.h>

// ---------------------------------------------------------------------------
// ComplexSVDTransform for MI455X (gfx1250, wave32, WMMA + TDM)
//
// y = x @ W @ W_inv  ==  x @ M   with  M = DFT . (I+P) . Neumann(P) . DFT^-1
// (alpha_scale cancels algebraically).  x: 65536x256 f32, M: 256x256 c64.
// Memory-bound: ~192MB HBM traffic -> ~8.3us floor @ 23.3 TB/s.
// ---------------------------------------------------------------------------

typedef __attribute__((ext_vector_type(2))) float v2f;
typedef __attribute__((ext_vector_type(8))) float v8f;
typedef unsigned int uint32x4 __attribute__((ext_vector_type(4)));
typedef int int32x8 __attribute__((ext_vector_type(8)));
typedef int int32x4 __attribute__((ext_vector_type(4)));

#define DIM 256
#define PLANE (DIM * DIM)
#define ROWS_TOTAL 65536
#define BLOCK_ROWS 64
#define TILE_ELEMS (BLOCK_ROWS * DIM)   // 16384 f32 = 64KB

// ---- setup kernel 1: unitary DFT matrix (built in double, like reference) --
__global__ void k_dft(float* __restrict__ dre, float* __restrict__ dim_) {
    int n = blockIdx.x, k = threadIdx.x;
    int p = (n * k) & (DIM - 1);
    double ang = 6.283185307179586476925287 * (double)p / (double)DIM;
    dre[n * DIM + k] = (float)(cos(ang) * 0.0625);   // 1/sqrt(256) = 1/16
    dim_[n * DIM + k] = (float)(-sin(ang) * 0.0625);
}

// ---- setup kernel 2: raw = (tanh(w_re) + i tanh(w_im)) * dim^-0.5 ----------
__global__ void k_tanh(const float* __restrict__ wre, const float* __restrict__ wim,
                       float* __restrict__ rre, float* __restrict__ rim) {
    int idx = blockIdx.x * DIM + threadIdx.x;
    rre[idx] = tanhf(wre[idx]) * 0.0625f;
    rim[idx] = tanhf(wim[idx]) * 0.0625f;
}

// ---- setup kernel 3: power iteration for sigma_max, then perturb = f*raw ---
__global__ void k_power(const float* __restrict__ rre, const float* __restrict__ rim,
                        const float* __restrict__ alpha_logit,
                        float* __restrict__ pre, float* __restrict__ pim) {
    __shared__ float vr[DIM], vi[DIM], ur[DIM], ui[DIM], red[DIM];
    __shared__ float fscale;
    const int j = threadIdx.x;
    vr[j] = 0.0625f; vi[j] = 0.0f;
    __syncthreads();

    for (int it = 0; it < 24; ++it) {
        float sr = 0.f, si = 0.f;
        for (int k = 0; k < DIM; ++k) {
            float ar = rre[j * DIM + k], ai = rim[j * DIM + k];
            float br = vr[k], bi = vi[k];
            sr += ar * br - ai * bi;
            si += ar * bi + ai * br;
        }
        __syncthreads();
        ur[j] = sr; ui[j] = si;
        __syncthreads();
        float wr = 0.f, wi = 0.f;
        for (int i = 0; i < DIM; ++i) {
            float ar = rre[i * DIM + j], ai = rim[i * DIM + j];
            float cr = ur[i], ci = ui[i];
            wr += ar * cr + ai * ci;
            wi += ar * ci - ai * cr;
        }
        red[j] = wr * wr + wi * wi;
        __syncthreads();
        for (int s = 128; s > 0; s >>= 1) {
            if (j < s) red[j] += red[j + s];
            __syncthreads();
        }
        float nw = sqrtf(red[0]);
        __syncthreads();
        float inv = (nw > 0.f) ? 1.f / nw : 0.f;
        vr[j] = wr * inv; vi[j] = wi * inv;
        __syncthreads();
    }
    float sr = 0.f, si = 0.f;
    for (int k = 0; k < DIM; ++k) {
        float ar = rre[j * DIM + k], ai = rim[j * DIM + k];
        sr += ar * vr[k] - ai * vi[k];
        si += ar * vi[k] + ai * vr[k];
    }
    red[j] = sr * sr + si * si;
    __syncthreads();
    for (int s = 128; s > 0; s >>= 1) {
        if (j < s) red[j] += red[j + s];
        __syncthreads();
    }
    if (j == 0) {
        float sigma = sqrtf(red[0]);
        float a = 1.f / (1.f + expf(-alpha_logit[0]));
        float beta = 1.f - a;
        float sn = fmaxf(sigma, 1.1920929e-07f);
        float scale = fminf(0.5f / sn, 1.f);   // RHO_MAX = 0.5
        fscale = beta * scale;
    }
    __syncthreads();
    float f = fscale;
    for (int k = 0; k < DIM; ++k) {
        pre[k * DIM + j] = f * rre[k * DIM + j];
        pim[k * DIM + j] = f * rim[k * DIM + j];
    }
}

// ---- setup kernel 4: term = I, learned_inv = I, learned = I + P ------------
__global__ void k_init(const float* __restrict__ pre, const float* __restrict__ pim,
                       float* __restrict__ tre, float* __restrict__ tim,
                       float* __restrict__ lire, float* __restrict__ liim,
                       float* __restrict__ lre, float* __restrict__ lim) {
    int i = blockIdx.x, j = threadIdx.x;
    int idx = i * DIM + j;
    float d = (i == j) ? 1.f : 0.f;
    tre[idx] = d;  tim[idx] = 0.f;
    lire[idx] = d; liim[idx] = 0.f;
    lre[idx] = pre[idx] + d;
    lim[idx] = pim[idx];
}

// ---- setup kernel 5: generic 256x256 complex matmul ------------------------
__global__ void k_cmatmul(const float* __restrict__ Are, const float* __restrict__ Aim,
                          const float* __restrict__ Bre, const float* __restrict__ Bim,
                          float* __restrict__ Cre, float* __restrict__ Cim,
                          float* accRe, float* accIm,
                          float sign, int conjB) {
    int i = blockIdx.x, j = threadIdx.x;
    float bs = conjB ? -1.f : 1.f;
    float sr = 0.f, si = 0.f;
    for (int k = 0; k < DIM; ++k) {
        float ar = Are[i * DIM + k], ai = Aim[i * DIM + k];
        float br = Bre[k * DIM + j], bi = bs * Bim[k * DIM + j];
        sr += ar * br - ai * bi;
        si += ar * bi + ai * br;
    }
    sr *= sign; si *= sign;
    int idx = i * DIM + j;
    Cre[idx] = sr; Cim[idx] = si;
    if (accRe) { accRe[idx] += sr; accIm[idx] += si; }
}

// ---- setup kernel 6: pack M into WMMA B-fragment order ---------------------
// record (tile 0..15, chunk 0..63, lane 0..31) -> float4 {br0, br1, bi0, bi1}
// kb = 4*chunk + 2*(lane>>4) ; col = tile*16 + (lane&15)
__global__ void k_packB(const float* __restrict__ Mre, const float* __restrict__ Mim,
                        float4* __restrict__ Bp) {
    int idx = blockIdx.x * 256 + threadIdx.x;   // 0 .. 32767
    int tile = idx >> 11;
    int rem = idx & 2047;
    int c = rem >> 5;
    int lane = rem & 31;
    int kb = 4 * c + 2 * (lane >> 4);
    int col = tile * 16 + (lane & 15);
    float4 v;
    v.x = Mre[(kb + 0) * DIM + col];
    v.y = Mre[(kb + 1) * DIM + col];
    v.z = Mim[(kb + 0) * DIM + col];
    v.w = Mim[(kb + 1) * DIM + col];
    Bp[idx] = v;
}

// ---- main kernel: y = x @ M  via V_WMMA_F32_16X16X4_F32 + TDM --------------
// grid: ROWS_TOTAL/BLOCK_ROWS blocks, 256 threads (8 waves).
// x tile (64x256 f32, contiguous 64KB) DMAd to LDS by the Tensor Data Mover.
__launch_bounds__(256)
__global__ void k_gemm_wmma(const float* __restrict__ x,
                            const float4* __restrict__ Bp,
                            float* __restrict__ out) {
    __shared__ float xs[TILE_ELEMS];   // 64 KB at LDS offset 0

    const int tid = threadIdx.x;
    const int rowBase = blockIdx.x * BLOCK_ROWS;

    // --- TDM: 1D tile descriptor, global -> LDS, issued by wave 0 only ------
    if (tid < 32) {
        unsigned long long ga = (unsigned long long)(x + (size_t)rowBase * DIM);
        uint32x4 g0;
        g0[0] = 1u;                                   // count=1 (valid D#)
        g0[1] = 0u;                                   // lds_addr = 0 (xs base)
        g0[2] = (unsigned int)(ga & 0xFFFFFFFFu);     // global_addr[31:0]
        g0[3] = (unsigned int)((ga >> 32) & 0x1FFFFFFu) | 0x80000000u; // [56:32] | type=2
        int32x8 g1;
        g1[0] = 0x00020000;       // workgroup_mask=0, data_size=2 (4 bytes)
        g1[1] = 0x40000000;       // tensor_dim0[15:0]=16384 (bits 63:48)
        g1[2] = 0x00010000;       // tensor_dim0 hi=0, tensor_dim1=1
        g1[3] = 0x40000000;       // tile_dim0=16384 (bits 127:112)
        g1[4] = 0;                // tile_dim1=0 (unused), tile_dim2=0
        g1[5] = 16384;            // tensor_dim0_stride lo32
        g1[6] = 0;
        g1[7] = 0;                // tensor_dim1_stride=0
        int32x4 z4 = {0, 0, 0, 0};
        int32x8 z8 = {0, 0, 0, 0, 0, 0, 0, 0};
        __builtin_amdgcn_tensor_load_to_lds(g0, g1, z4, z4, z8, 0);
        __builtin_amdgcn_s_wait_tensorcnt(0);
    }
    __syncthreads();
    // The TDM builtin carries no pointer to xs, so clang's alias analysis would
    // otherwise fold all xs reads to undef (observed in round-3 asm: A-fragment
    // ds_loads deleted). Escape xs into a memory-clobbering asm so the compiler
    // must treat LDS as written by the DMA.
    asm volatile("" : : "v"(xs) : "memory");

    const int wave = tid >> 5;
    const int lane = tid & 31;
    const int strip = wave >> 1;      // row strip: rows strip*16 .. +15
    const int half = wave & 1;        // N half: tiles half*8 .. half*8+7
    const int laneHalf = lane >> 4;   // 0/1 (K sub-group for A/B fragments)
    const int lmod = lane & 15;

    // A fragment base: row = strip*16 + lmod, K offset 2*laneHalf (ISA 7.12.2)
    const float* arow = xs + (strip * 16 + lmod) * DIM + 2 * laneHalf;

    float2* o2 = (float2*)out;

    for (int t = 0; t < 8; ++t) {
        const int tile = half * 8 + t;
        const int n0 = tile * 16;
        const float4* bptr = Bp + ((size_t)tile * 64) * 32 + lane;
        v8f accr = {};
        v8f acci = {};
        for (int c = 0; c < 64; ++c) {
            v2f a = *(const v2f*)(arow + 4 * c);      // ds_load_b64
            float4 b = bptr[(size_t)c * 32];          // coalesced global_load_b128
            v2f br; br[0] = b.x; br[1] = b.y;
            v2f bi; bi[0] = b.z; bi[1] = b.w;
            accr = __builtin_amdgcn_wmma_f32_16x16x4_f32(
                false, a, false, br, (short)0, accr, false, false);
            acci = __builtin_amdgcn_wmma_f32_16x16x4_f32(
                false, a, false, bi, (short)0, acci, false, false);
        }
        // C/D layout: VGPR i -> M = i + 8*laneHalf, N = lmod. Store interleaved c64.
#pragma unroll
        for (int i = 0; i < 8; ++i) {
            int r = rowBase + strip * 16 + i + 8 * laneHalf;
            int c = n0 + lmod;
            float2 v;
            v.x = accr[i];
            v.y = acci[i];
            o2[(size_t)r * DIM + c] = v;              // global_store_b64
        }
    }
}

// ---------------------------------------------------------------------------
extern "C" void kernel_launch(void* const* d_in, const int* in_sizes, int n_in,
                              void* d_out, int out_size, void* d_ws, size_t ws_size,
                              hipStream_t stream) {
    const float* x = (const float*)d_in[0];
    const float* w_re = (const float*)d_in[1];
    const float* w_im = (const float*)d_in[2];
    const float* alpha_logit = (const float*)d_in[3];
    float* out = (float*)d_out;
    float* ws = (float*)d_ws;

    // workspace planes (22 x 256KB = 5.5MB)
    float* dre  = ws + 0 * PLANE;  float* dimg = ws + 1 * PLANE;   // DFT
    float* rre  = ws + 2 * PLANE;  float* rim  = ws + 3 * PLANE;   // raw
    float* pre  = ws + 4 * PLANE;  float* pim  = ws + 5 * PLANE;   // perturb
    float* tAre = ws + 6 * PLANE;  float* tAim = ws + 7 * PLANE;   // term ping
    float* tBre = ws + 8 * PLANE;  float* tBim = ws + 9 * PLANE;   // term pong
    float* lire = ws + 10 * PLANE; float* liim = ws + 11 * PLANE;  // learned_inv
    float* lre  = ws + 12 * PLANE; float* lim  = ws + 13 * PLANE;  // learned
    float* t1re = ws + 14 * PLANE; float* t1im = ws + 15 * PLANE;
    float* t2re = ws + 16 * PLANE; float* t2im = ws + 17 * PLANE;
    float* mre  = ws + 18 * PLANE; float* mim  = ws + 19 * PLANE;  // final M
    float4* bp  = (float4*)(ws + 20 * PLANE);                      // packed B (512KB)

    k_dft<<<DIM, DIM, 0, stream>>>(dre, dimg);
    k_tanh<<<DIM, DIM, 0, stream>>>(w_re, w_im, rre, rim);
    k_power<<<1, DIM, 0, stream>>>(rre, rim, alpha_logit, pre, pim);
    k_init<<<DIM, DIM, 0, stream>>>(pre, pim, tAre, tAim, lire, liim, lre, lim);

    // Neumann: term = -(term @ P); learned_inv += term   (8 terms)
    float* sre = tAre; float* sim = tAim;
    float* qre = tBre; float* qim = tBim;
    for (int t = 0; t < 8; ++t) {
        k_cmatmul<<<DIM, DIM, 0, stream>>>(sre, sim, pre, pim, qre, qim,
                                           lire, liim, -1.f, 0);
        float* tr = sre; sre = qre; qre = tr;
        float* ti = sim; sim = qim; qim = ti;
    }

    // M = DFT @ (learned @ learned_inv) @ DFT^-1   (DFT^-1 = conj(DFT), symmetric)
    k_cmatmul<<<DIM, DIM, 0, stream>>>(lre, lim, lire, liim, t1re, t1im,
                                       nullptr, nullptr, 1.f, 0);
    k_cmatmul<<<DIM, DIM, 0, stream>>>(t1re, t1im, dre, dimg, t2re, t2im,
                                       nullptr, nullptr, 1.f, 1);
    k_cmatmul<<<DIM, DIM, 0, stream>>>(dre, dimg, t2re, t2im, mre, mim,
                                       nullptr, nullptr, 1.f, 0);

    // pack M into WMMA fragment order (one b128 per lane per k-chunk)
    k_packB<<<128, 256, 0, stream>>>(mre, mim, bp);

    // main WMMA GEMM: y = x @ M
    k_gemm_wmma<<<ROWS_TOTAL / BLOCK_ROWS, 256, 0, stream>>>(x, bp, out);
}